// BilateralGridEditor_9878424780966
// MI455X (gfx1250) — compile-verified
//
#include <hip/hip_runtime.h>
#include <math.h>

// ---------------------------------------------------------------------------
// Fused white-balance (Bradford) + image statistics for MI455X (gfx1250).
//  - single streaming pass over the 96MB image (read once, write once)
//  - per-batch 3x3 color matrix computed with one V_WMMA_F32_16X16X4_F32
//  - deterministic reductions (shfl_xor -> LDS -> per-block partials ->
//    fixed-order sum); histograms use integer atomics (order-independent)
// ---------------------------------------------------------------------------

static constexpr int IMG_H  = 1024;
static constexpr int IMG_W  = 1024;
static constexpr int IMG_HW = IMG_H * IMG_W;       // 1048576 pixels / batch
static constexpr int NBINS  = 32768;               // fine luma histogram bins
static constexpr int NACC   = 52;                  // streaming accumulators
static constexpr int BLOCKS_PER_BATCH = 256;
static constexpr int CENTER_N = (IMG_H / 2) * (IMG_W / 2); // 262144

typedef float v2f __attribute__((ext_vector_type(2)));
typedef float v8f __attribute__((ext_vector_type(8)));

__device__ __forceinline__ float clampf(float x, float lo, float hi) {
    return fminf(fmaxf(x, lo), hi);
}
__device__ __forceinline__ float fpow(float x, float p) {
    return exp2f(p * log2f(x));   // lowers to v_log_f32 / v_exp_f32
}

// ---------------------------------------------------------------------------
// Kernel 0: zero the fine histograms (harness poisons ws with 0xAA).
// ---------------------------------------------------------------------------
__global__ void zero_u32(unsigned* __restrict__ p, int n) {
    for (int i = blockIdx.x * blockDim.x + threadIdx.x; i < n;
         i += gridDim.x * blockDim.x)
        p[i] = 0u;
}

// ---------------------------------------------------------------------------
// Kernel 1: per-batch Bradford chromatic-adaptation matrix.
// M_total = P * diag(D) * Q with P = XYZ2RGB*BRADFORD_INV, Q = BRADFORD*RGB2XYZ
// -> a single 3x3 matmul per batch, executed as V_WMMA_F32_16X16X4_F32.
// One wave (32 threads) per batch; EXEC is all-ones as WMMA requires.
// ---------------------------------------------------------------------------
__global__ __launch_bounds__(32) void wb_setup(const float* __restrict__ temp,
                                               const float* __restrict__ tint,
                                               float* __restrict__ Mt) {
    const int b    = blockIdx.x;
    const int lane = threadIdx.x;

    const float BR[3][3]  = {{0.8951f, 0.2664f, -0.1614f},
                             {-0.7502f, 1.7135f, 0.0367f},
                             {0.0389f, -0.0685f, 1.0296f}};
    const float BRI[3][3] = {{0.9869929f, -0.1470543f, 0.1599627f},
                             {0.4323053f, 0.5183603f, 0.0492912f},
                             {-0.0085287f, 0.0400428f, 0.9684867f}};
    const float R2X[3][3] = {{0.4124564f, 0.3575761f, 0.1804375f},
                             {0.2126729f, 0.7151522f, 0.072175f},
                             {0.0193339f, 0.119192f, 0.9503041f}};
    const float X2R[3][3] = {{3.2404542f, -1.5371385f, -0.4985314f},
                             {-0.969266f, 1.8760108f, 0.041556f},
                             {0.0556434f, -0.2040259f, 1.0572252f}};

    // --- temp/tint -> destination white point (uniform across lanes) ---
    float t = clampf(temp[b], -1.f, 1.f);
    float k = clampf(tint[b], -1.f, 1.f);
    float base_mired = 1000000.f / 6500.f;
    float cct = clampf(1000000.f / fmaxf(base_mired + t * 120.f, 1.f),
                       2000.f, 20000.f);
    float T  = clampf(cct, 1667.f, 25000.f);
    float iT = 1.f / T;
    float x;
    if (T <= 4000.f)
        x = ((-2.661239e8f * iT - 2.34358e5f) * iT + 877.6956f) * iT + 0.17991f;
    else
        x = ((-3.0258469e9f * iT + 2.1070379e6f) * iT + 222.6347f) * iT + 0.24039f;
    float y;
    if (T <= 2222.f)
        y = ((-1.1063814f * x - 1.3481102f) * x + 2.18555832f) * x - 0.20219683f;
    else if (T <= 4000.f)
        y = ((-0.9549476f * x - 1.37418593f) * x + 2.09137015f) * x - 0.16748867f;
    else
        y = ((3.081758f * x - 5.8733867f) * x + 3.75112997f) * x - 0.37001483f;
    x = clampf(x, 0.f, 0.99f);
    y = clampf(y, 0.f, 0.99f);
    float den = fmaxf(-2.f * x + 12.f * y + 3.f, 1e-6f);
    float u   = 4.f * x / den;
    float v   = 6.f * y / den + k * 0.06f;
    float den2 = fmaxf(2.f * u - 8.f * v + 4.f, 1e-6f);
    float xt = clampf(3.f * u / den2, 0.f, 0.99f);
    float yt = clampf(2.f * v / den2, 0.f, 0.99f);

    float sx = 0.3127f, sy = 0.329f;
    float srcXYZ[3] = {sx / sy, 1.f, (1.f - sx - sy) / sy};
    float dx = clampf(xt, 1e-6f, 1.f), dy = clampf(yt, 1e-6f, 1.f);
    float dstXYZ[3] = {dx / dy, 1.f, (1.f - dx - dy) / dy};

    float Dv3[3];
    for (int i = 0; i < 3; ++i) {
        float sL = BR[i][0] * srcXYZ[0] + BR[i][1] * srcXYZ[1] + BR[i][2] * srcXYZ[2];
        float dL = BR[i][0] * dstXYZ[0] + BR[i][1] * dstXYZ[1] + BR[i][2] * dstXYZ[2];
        Dv3[i] = clampf(dL / (sL + 1e-6f), 0.25f, 4.f);
    }

    // P = X2R*BRI, Q = BR*R2X (3x3 each)
    float P[3][3], Q[3][3];
    for (int i = 0; i < 3; ++i)
        for (int j = 0; j < 3; ++j) {
            float s1 = 0.f, s2 = 0.f;
            for (int kk = 0; kk < 3; ++kk) {
                s1 += X2R[i][kk] * BRI[kk][j];
                s2 += BR[i][kk] * R2X[kk][j];
            }
            P[i][j] = s1;
            Q[i][j] = s2;
        }

    // --- pack fragments for V_WMMA_F32_16X16X4_F32 (wave32) ---
    // A (16x4): vgpr0 = K0 (lanes 0-15) / K2 (lanes 16-31), vgpr1 = K1 / K3.
    // B (4x16): lane<16 holds col=lane K0/K1, lane>=16 holds col=lane-16 K2/K3.
    const int  row = lane & 15;
    const bool hiH = lane >= 16;
    float ax = 0.f, ay = 0.f, bx = 0.f, by = 0.f;
    if (row < 3) {
        ax = hiH ? P[row][2] : P[row][0];
        ay = hiH ? 0.f       : P[row][1];
        bx = hiH ? Dv3[2] * Q[2][row] : Dv3[0] * Q[0][row];
        by = hiH ? 0.f                : Dv3[1] * Q[1][row];
    }
    v2f Af; Af.x = ax; Af.y = ay;
    v2f Bf; Bf.x = bx; Bf.y = by;
    v8f Cf = {0.f, 0.f, 0.f, 0.f, 0.f, 0.f, 0.f, 0.f};
    v8f Df = __builtin_amdgcn_wmma_f32_16x16x4_f32(false, Af, false, Bf,
                                                   (short)0, Cf, false, false);
    // D layout: vgpr m holds row m (lanes 0-15); M_total rows 0..2, cols 0..2.
    if (lane < 3) {
        Mt[b * 9 + 0 + lane] = Df[0];
        Mt[b * 9 + 3 + lane] = Df[1];
        Mt[b * 9 + 6 + lane] = Df[2];
    }
}

// ---------------------------------------------------------------------------
// Kernel 2: fused transform + statistics streaming pass.
// grid = (BLOCKS_PER_BATCH, B), block = 256 (8 wave32 waves).
// ---------------------------------------------------------------------------
__global__ __launch_bounds__(256) void wb_main(
    const float* __restrict__ rgb, const float* __restrict__ Mt,
    float* __restrict__ out, float* __restrict__ partials,
    unsigned* __restrict__ histM, unsigned* __restrict__ histC) {
    const int    bat  = blockIdx.y;
    const size_t base = (size_t)bat * 3u * IMG_HW;
    const float* Mr   = Mt + bat * 9;
    const float m00 = Mr[0], m01 = Mr[1], m02 = Mr[2];
    const float m10 = Mr[3], m11 = Mr[4], m12 = Mr[5];
    const float m20 = Mr[6], m21 = Mr[7], m22 = Mr[8];
    unsigned* hM = histM + (size_t)bat * NBINS;
    unsigned* hC = histC + (size_t)bat * NBINS;

    float acc[NACC];
#pragma unroll
    for (int i = 0; i < NACC; ++i) acc[i] = 0.f;

    const int nq = IMG_HW / 4;
    for (int q = blockIdx.x * blockDim.x + threadIdx.x; q < nq;
         q += gridDim.x * blockDim.x) {
        const int p = q * 4;
        float4 r4 = *(const float4*)(rgb + base + p);
        float4 g4 = *(const float4*)(rgb + base + IMG_HW + p);
        float4 b4 = *(const float4*)(rgb + base + 2 * IMG_HW + p);
        float ri[4] = {r4.x, r4.y, r4.z, r4.w};
        float gi[4] = {g4.x, g4.y, g4.z, g4.w};
        float bi[4] = {b4.x, b4.y, b4.z, b4.w};
        float ro[4], go[4], bo[4];
#pragma unroll
        for (int j = 0; j < 4; ++j) {
            // srgb -> linear
            float sr = clampf(ri[j], 0.f, 1.f);
            float sg = clampf(gi[j], 0.f, 1.f);
            float sb = clampf(bi[j], 0.f, 1.f);
            float lr = sr <= 0.04045f ? sr * (1.f / 12.92f)
                                      : fpow((sr + 0.055f) * (1.f / 1.055f), 2.4f);
            float lg = sg <= 0.04045f ? sg * (1.f / 12.92f)
                                      : fpow((sg + 0.055f) * (1.f / 1.055f), 2.4f);
            float lb = sb <= 0.04045f ? sb * (1.f / 12.92f)
                                      : fpow((sb + 0.055f) * (1.f / 1.055f), 2.4f);
            float luma0 = 0.2126f * lr + 0.7152f * lg + 0.0722f * lb;
            // 3x3 transform
            float tr = m00 * lr + m01 * lg + m02 * lb;
            float tg = m10 * lr + m11 * lg + m12 * lb;
            float tb = m20 * lr + m21 * lg + m22 * lb;
            float luma1 = 0.2126f * tr + 0.7152f * tg + 0.0722f * tb;
            float sc = luma0 / (luma1 + 1e-6f);
            tr = clampf(tr * sc, 0.f, 1.f);
            tg = clampf(tg * sc, 0.f, 1.f);
            tb = clampf(tb * sc, 0.f, 1.f);
            // linear -> srgb
            float R = tr <= 0.0031308f
                          ? tr * 12.92f
                          : 1.055f * fpow(fmaxf(tr, 1e-5f), 1.f / 2.4f) - 0.055f;
            float G = tg <= 0.0031308f
                          ? tg * 12.92f
                          : 1.055f * fpow(fmaxf(tg, 1e-5f), 1.f / 2.4f) - 0.055f;
            float B = tb <= 0.0031308f
                          ? tb * 12.92f
                          : 1.055f * fpow(fmaxf(tb, 1e-5f), 1.f / 2.4f) - 0.055f;
            R = clampf(R, 0.f, 1.f);
            G = clampf(G, 0.f, 1.f);
            B = clampf(B, 0.f, 1.f);
            ro[j] = R; go[j] = G; bo[j] = B;

            // ---------------- statistics on the output ----------------
            float l = 0.299f * R + 0.587f * G + 0.114f * B;
            acc[0] += l;  acc[1] += l * l;
            acc[2] += R;  acc[3] += G;  acc[4] += B;
            float u = -0.14713f * R - 0.28886f * G + 0.436f * B;
            float v =  0.615f   * R - 0.51499f * G - 0.10001f * B;
            acc[5] += u; acc[6] += u * u; acc[7] += v; acc[8] += v * v;
            float ch = sqrtf(u * u + v * v + 1e-8f);
            bool shB = l < 0.25f, hiB = l >= 0.75f;
            bool mdB = !shB && !hiB;
            acc[9]  += shB ? 1.f : 0.f;
            acc[10] += mdB ? 1.f : 0.f;
            acc[11] += hiB ? 1.f : 0.f;
            acc[12] += shB ? ch : 0.f;
            acc[13] += mdB ? ch : 0.f;
            acc[14] += hiB ? ch : 0.f;
            acc[15] += fmaxf(R - 0.98f, 0.f);
            acc[16] += fmaxf(G - 0.98f, 0.f);
            acc[17] += fmaxf(B - 0.98f, 0.f);
            acc[18] += fmaxf(0.02f - R, 0.f);
            acc[19] += fmaxf(0.02f - G, 0.f);
            acc[20] += fmaxf(0.02f - B, 0.f);

            int pix = p + j;
            int row = pix >> 10, col = pix & 1023;
            bool ctr = (row >= 256) && (row < 768) && (col >= 256) && (col < 768);
            int fb = min((int)(l * (float)NBINS), NBINS - 1);
            fb = max(fb, 0);
            atomicAdd(&hM[fb], 1u);
            if (ctr) {
                acc[21] += l;
                acc[22] += l * l;
                acc[23] += ((R > 0.995f) || (G > 0.995f) || (B > 0.995f)) ? 1.f : 0.f;
                atomicAdd(&hC[fb], 1u);
            }
            float hue = atan2f(v, u);
            float hn  = hue * 0.15915494309189535f + 0.5f;
            hn -= floorf(hn);
            int hb = min((int)(hn * 12.f), 11); hb = max(hb, 0);
#pragma unroll
            for (int i = 0; i < 12; ++i) acc[24 + i] += (hb == i) ? ch : 0.f;
            int i16 = min((int)(l * 15.f), 15); i16 = max(i16, 0);
#pragma unroll
            for (int i = 0; i < 16; ++i) acc[36 + i] += (i16 == i) ? 1.f : 0.f;
        }
        *(float4*)(out + base + p)              = make_float4(ro[0], ro[1], ro[2], ro[3]);
        *(float4*)(out + base + IMG_HW + p)     = make_float4(go[0], go[1], go[2], go[3]);
        *(float4*)(out + base + 2 * IMG_HW + p) = make_float4(bo[0], bo[1], bo[2], bo[3]);
    }

    // deterministic block reduction: wave32 butterfly -> LDS -> fixed-order sum
    __shared__ float red[8][NACC];
    const int wid = threadIdx.x >> 5, lane = threadIdx.x & 31;
#pragma unroll
    for (int i = 0; i < NACC; ++i) {
        float v = acc[i];
        for (int off = 16; off > 0; off >>= 1) v += __shfl_xor(v, off, 32);
        if (lane == 0) red[wid][i] = v;
    }
    __syncthreads();
    for (int i = threadIdx.x; i < NACC; i += blockDim.x) {
        float s = 0.f;
        for (int w = 0; w < 8; ++w) s += red[w][i];
        partials[((size_t)bat * gridDim.x + blockIdx.x) * NACC + i] = s;
    }
}

// ---------------------------------------------------------------------------
// Order statistics from fine histogram (cooperative scan, 256 threads).
// ---------------------------------------------------------------------------
__device__ void hist_orderstats(const unsigned* __restrict__ h, const int* tk,
                                float* tv, int nT, unsigned* seg) {
    const int t     = threadIdx.x;
    const int span  = NBINS / 256;
    const int sbase = t * span;
    unsigned s = 0;
    for (int i = 0; i < span; ++i) s += h[sbase + i];
    seg[t] = s;
    __syncthreads();
    if (t == 0) {
        unsigned c = 0;
        for (int i = 0; i < 256; ++i) { unsigned x = seg[i]; seg[i] = c; c += x; }
    }
    __syncthreads();
    unsigned cum = seg[t];
    for (int i = 0; i < span; ++i) {
        unsigned c = h[sbase + i];
        if (c) {
            for (int j = 0; j < nT; ++j) {
                int k = tk[j];
                if (k >= (int)cum && k < (int)(cum + c))
                    tv[j] = ((float)(sbase + i) +
                             ((float)(k - (int)cum) + 0.5f) / (float)c) *
                            (1.f / (float)NBINS);
            }
            cum += c;
        }
    }
    __syncthreads();
}

// ---------------------------------------------------------------------------
// Kernel 3: finalize the 61 statistics per batch (one block per batch).
// ---------------------------------------------------------------------------
__global__ __launch_bounds__(256) void wb_finalize(
    const float* __restrict__ partials, const unsigned* __restrict__ histM,
    const unsigned* __restrict__ histC, float* __restrict__ stats, int nblocks) {
    const int bat = blockIdx.x;
    __shared__ float    S[NACC];
    __shared__ unsigned seg[256];
    __shared__ int      tk[18];
    __shared__ float    tv[18];
    __shared__ float    qmain[9], qcent[3];

    for (int i = threadIdx.x; i < NACC; i += blockDim.x) {
        float s = 0.f;
        const float* pp = partials + ((size_t)bat * nblocks) * NACC + i;
        for (int bl = 0; bl < nblocks; ++bl) s += pp[(size_t)bl * NACC];
        S[i] = s;
    }
    __syncthreads();

    const float Nf  = (float)IMG_HW;
    const float Ncf = (float)CENTER_N;
    const float qlev[9] = {0.01f, 0.05f, 0.1f, 0.25f, 0.5f, 0.75f, 0.9f, 0.95f, 0.99f};
    const float clev[3] = {0.1f, 0.5f, 0.9f};

    // main-luma quantiles
    if (threadIdx.x == 0) {
        for (int j = 0; j < 9; ++j) {
            float hh = qlev[j] * (Nf - 1.f);
            int klo  = (int)floorf(hh);
            tk[2 * j]     = klo;
            tk[2 * j + 1] = min(klo + 1, IMG_HW - 1);
        }
        for (int j = 0; j < 18; ++j) tv[j] = 0.f;
    }
    __syncthreads();
    hist_orderstats(histM + (size_t)bat * NBINS, tk, tv, 18, seg);
    if (threadIdx.x == 0)
        for (int j = 0; j < 9; ++j) {
            float hh = qlev[j] * (Nf - 1.f);
            float fr = hh - floorf(hh);
            qmain[j] = tv[2 * j] + fr * (tv[2 * j + 1] - tv[2 * j]);
        }
    __syncthreads();

    // center-luma quantiles
    if (threadIdx.x == 0) {
        for (int j = 0; j < 3; ++j) {
            float hh = clev[j] * (Ncf - 1.f);
            int klo  = (int)floorf(hh);
            tk[2 * j]     = klo;
            tk[2 * j + 1] = min(klo + 1, CENTER_N - 1);
        }
        for (int j = 0; j < 6; ++j) tv[j] = 0.f;
    }
    __syncthreads();
    hist_orderstats(histC + (size_t)bat * NBINS, tk, tv, 6, seg);
    if (threadIdx.x == 0)
        for (int j = 0; j < 3; ++j) {
            float hh = clev[j] * (Ncf - 1.f);
            float fr = hh - floorf(hh);
            qcent[j] = tv[2 * j] + fr * (tv[2 * j + 1] - tv[2 * j]);
        }
    __syncthreads();

    if (threadIdx.x == 0) {
        float* o = stats + (size_t)bat * 61;
        o[0] = S[0] / Nf;
        float var = (S[1] - S[0] * S[0] / Nf) / (Nf - 1.f);
        o[1] = clampf(sqrtf(fmaxf(var, 0.f)), 0.f, 0.5f) * 2.f;
        for (int j = 0; j < 9; ++j) o[2 + j] = qmain[j];
        for (int i = 0; i < 16; ++i) o[11 + i] = S[36 + i] / (Nf + 1e-6f);
        o[27] = S[12] / fmaxf(S[9], 1.f);
        o[28] = S[13] / fmaxf(S[10], 1.f);
        o[29] = S[14] / fmaxf(S[11], 1.f);
        o[30] = S[9] / Nf;  o[31] = S[10] / Nf; o[32] = S[11] / Nf;
        o[33] = S[15] / Nf; o[34] = S[16] / Nf; o[35] = S[17] / Nf;
        o[36] = S[18] / Nf; o[37] = S[19] / Nf; o[38] = S[20] / Nf;
        float rm = fmaxf(S[2] / Nf, 1e-6f);
        float gm = fmaxf(S[3] / Nf, 1e-6f);
        float bm = fmaxf(S[4] / Nf, 1e-6f);
        o[39] = logf(rm / gm);
        o[40] = logf(bm / gm);
        float uvar = (S[6] - S[5] * S[5] / Nf) / (Nf - 1.f);
        o[41] = clampf(sqrtf(fmaxf(uvar, 0.f)), 0.f, 0.25f) * 4.f;
        float vvar = (S[8] - S[7] * S[7] / Nf) / (Nf - 1.f);
        o[42] = clampf(sqrtf(fmaxf(vvar, 0.f)), 0.f, 0.25f) * 4.f;
        o[43] = S[21] / Ncf;
        float cvar = (S[22] - S[21] * S[21] / Ncf) / (Ncf - 1.f);
        o[44] = clampf(sqrtf(fmaxf(cvar, 0.f)), 0.f, 0.5f) * 2.f;
        o[45] = qcent[0]; o[46] = qcent[1]; o[47] = qcent[2];
        o[48] = S[23] / Ncf;
        for (int i = 0; i < 12; ++i)
            o[49 + i] = S[24 + i] / (Nf + 1e-6f) * 10.f;
    }
}

// ---------------------------------------------------------------------------
extern "C" void kernel_launch(void* const* d_in, const int* in_sizes, int n_in,
                              void* d_out, int out_size, void* d_ws, size_t ws_size,
                              hipStream_t stream) {
    const float* rgb  = (const float*)d_in[0];
    const float* temp = (const float*)d_in[1];
    const float* tint = (const float*)d_in[2];
    const int B = in_sizes[1];                 // batch (8)
    float* out   = (float*)d_out;
    float* stats = out + (size_t)in_sizes[0];  // stats follow the image

    // workspace layout
    char*  ws  = (char*)d_ws;
    size_t off = 0;
    float* partials = (float*)(ws + off);
    off += (size_t)B * BLOCKS_PER_BATCH * NACC * sizeof(float);
    unsigned* histM = (unsigned*)(ws + off);
    off += (size_t)B * NBINS * sizeof(unsigned);
    unsigned* histC = (unsigned*)(ws + off);
    off += (size_t)B * NBINS * sizeof(unsigned);
    float* Mt = (float*)(ws + off);
    off += (size_t)B * 9 * sizeof(float);
    (void)ws_size; (void)n_in; (void)out_size;

    const int zwords = 2 * B * NBINS;          // histM and histC are contiguous
    zero_u32<<<1024, 256, 0, stream>>>(histM, zwords);
    wb_setup<<<B, 32, 0, stream>>>(temp, tint, Mt);
    wb_main<<<dim3(BLOCKS_PER_BATCH, B), 256, 0, stream>>>(rgb, Mt, out, partials,
                                                           histM, histC);
    wb_finalize<<<B, 256, 0, stream>>>(partials, histM, histC, stats,
                                       BLOCKS_PER_BATCH);
}